// ChebNet_82197084111197
// MI455X (gfx1250) — compile-verified
//
#include <hip/hip_runtime.h>
#include <cmath>

#define N_NODES 50000
#define N_EDGES 800000
#define HID     96
#define NGRAPH  128
#define NCLS    10
#define BN_EPS  1e-5f

typedef __attribute__((ext_vector_type(2))) float v2f;
typedef __attribute__((ext_vector_type(8))) float v8f;

// ---------------- elementwise helpers ----------------

__global__ void fill_kernel(float* __restrict__ p, int n, float v) {
    int i = blockIdx.x * blockDim.x + threadIdx.x;
    if (i < n) p[i] = v;
}

__global__ void deg_kernel(const int* __restrict__ dst, float* __restrict__ deg, int e) {
    int i = blockIdx.x * blockDim.x + threadIdx.x;
    if (i < e) atomicAdd(&deg[dst[i]], 1.0f);
}

__global__ void norm_kernel(const int* __restrict__ src, const int* __restrict__ dst,
                            const float* __restrict__ deg, float* __restrict__ nrm, int e) {
    int i = blockIdx.x * blockDim.x + threadIdx.x;
    if (i < e) {
        float ds = deg[src[i]], dd = deg[dst[i]];
        float is = ds > 0.f ? rsqrtf(fmaxf(ds, 1.f)) : 0.f;
        float id = dd > 0.f ? rsqrtf(fmaxf(dd, 1.f)) : 0.f;
        nrm[i] = -(is * id);
    }
}

// out[dst] += norm[e] * h[src]; 4 features per thread
__global__ void scatter_kernel(const int* __restrict__ src, const int* __restrict__ dst,
                               const float* __restrict__ nrm, const float* __restrict__ h,
                               float* __restrict__ out, int e) {
    int idx = blockIdx.x * blockDim.x + threadIdx.x;
    if (idx >= e * 24) return;
    int ed = idx / 24;
    int f0 = (idx % 24) * 4;
    int s = src[ed], d = dst[ed];
    float w = nrm[ed];
    const float* hp = h + (size_t)s * HID + f0;
    float* op = out + (size_t)d * HID + f0;
    atomicAdd(op + 0, w * hp[0]);
    atomicAdd(op + 1, w * hp[1]);
    atomicAdd(op + 2, w * hp[2]);
    atomicAdd(op + 3, w * hp[3]);
}

// Cheb recurrence: t = 2*t - s
__global__ void recur_kernel(float* __restrict__ t, const float* __restrict__ s, int n) {
    int i = blockIdx.x * blockDim.x + threadIdx.x;
    if (i < n) t[i] = 2.0f * t[i] - s[i];
}

// d = a + b (residual)
__global__ void add_kernel(float* __restrict__ d, const float* __restrict__ a,
                           const float* __restrict__ b, int n) {
    int i = blockIdx.x * blockDim.x + threadIdx.x;
    if (i < n) d[i] = a[i] + b[i];
}

// ---------------- WMMA GEMM: out[rows,96] = sum_k A_k[rows,KDIM] @ W[k][KDIM,96]
// then +bias, BatchNorm(eval), LeakyReLU(alpha) (alpha=0 -> ReLU).
// One wave computes RB row-tiles (RB*16 rows) x 16 cols: B loaded once per K-step,
// reused across RB WMMAs -> RB independent accumulator chains hide WMMA RAW latency.
// blockDim=(32,6) covers all 96 output columns.

template <int NK, int KDIM, int RB>
__global__ __launch_bounds__(192) void gemm_bn_act_kernel(
    const float* __restrict__ A0, const float* __restrict__ A1,
    const float* __restrict__ A2, const float* __restrict__ A3,
    const float* __restrict__ W,  const float* __restrict__ bias,
    const float* __restrict__ mean, const float* __restrict__ var,
    const float* __restrict__ gamma, const float* __restrict__ beta,
    float alpha, float* __restrict__ out)
{
    const float* As[4] = {A0, A1, A2, A3};
    const int lane = threadIdx.x;      // 0..31
    const int m    = lane & 15;        // row-in-tile (A) / col-in-tile (B,C)
    const int hi   = lane >> 4;        // K-half selector per ISA 16x4 f32 layout
    const int row0 = blockIdx.x * (16 * RB);
    const int col0 = threadIdx.y * 16;

    v8f acc[RB];
#pragma unroll
    for (int r = 0; r < RB; ++r)
        acc[r] = (v8f){0.f, 0.f, 0.f, 0.f, 0.f, 0.f, 0.f, 0.f};

#pragma unroll
    for (int k = 0; k < NK; ++k) {
        const float* abase = As[k] + (size_t)(row0 + m) * KDIM + hi * 2;
        const float* Wk    = W + (size_t)k * KDIM * HID;
#pragma unroll
        for (int kk = 0; kk < KDIM; kk += 4) {
            // B 4x16: v0 lanes carry rows K=kk+hi*2, v1 rows K=kk+hi*2+1
            v2f b;
            const float* wp = Wk + (size_t)(kk + hi * 2) * HID + col0 + m;
            b.x = wp[0];
            b.y = wp[HID];
#pragma unroll
            for (int r = 0; r < RB; ++r) {
                // A 16x4: lanes 0-15 -> K=kk+0,kk+1 ; lanes 16-31 -> K=kk+2,kk+3
                v2f a;
                const float* ap = abase + (size_t)(r * 16) * KDIM + kk;
                a.x = ap[0];
                a.y = ap[1];
                acc[r] = __builtin_amdgcn_wmma_f32_16x16x4_f32(
                    false, a, false, b, (short)0, acc[r], false, false);
            }
        }
    }

    // Epilogue: C/D layout — VGPR i: lanes 0-15 -> M=i, lanes 16-31 -> M=i+8; N=lane%16
    const int col = col0 + m;
    const float scale = gamma[col] * rsqrtf(var[col] + BN_EPS);
    const float shift = beta[col] - mean[col] * scale;
    const float bz = bias[col];
#pragma unroll
    for (int r = 0; r < RB; ++r) {
#pragma unroll
        for (int i = 0; i < 8; ++i) {
            const int row = row0 + r * 16 + i + hi * 8;
            float v = (acc[r][i] + bz) * scale + shift;
            v = v > 0.f ? v : alpha * v;
            out[(size_t)row * HID + col] = v;
        }
    }
}

// ---------------- segment-max pool over concat [xt,x1,x2,x3] ----------------

__device__ inline void atomicMaxF(float* addr, float v) {
    if (v >= 0.f) atomicMax((int*)addr, __float_as_int(v));
    else          atomicMin((unsigned int*)addr, __float_as_uint(v));
}

__global__ void pool_kernel(const float* __restrict__ xt, const float* __restrict__ x1,
                            const float* __restrict__ x2, const float* __restrict__ x3,
                            const int* __restrict__ batch, float* __restrict__ pooled) {
    int idx = blockIdx.x * blockDim.x + threadIdx.x;
    if (idx >= N_NODES * 4 * HID) return;
    int node = idx / (4 * HID);
    int f = idx % (4 * HID);
    const float* srcs[4] = {xt, x1, x2, x3};
    float v = srcs[f / HID][(size_t)node * HID + (f % HID)];
    int g = batch[node];
    atomicMaxF(&pooled[(size_t)g * (4 * HID) + f], v);
}

// ---------------- final tiny GEMM 128x96 @ 96x10 ----------------

__global__ void mlp2_kernel(const float* __restrict__ z1, const float* __restrict__ W2,
                            const float* __restrict__ b2, float* __restrict__ out) {
    int idx = blockIdx.x * blockDim.x + threadIdx.x;
    if (idx >= NGRAPH * NCLS) return;
    int g = idx / NCLS, c = idx % NCLS;
    float s = b2[c];
#pragma unroll 8
    for (int j = 0; j < HID; ++j) s += z1[(size_t)g * HID + j] * W2[(size_t)j * NCLS + c];
    out[idx] = s;
}

// ---------------- host orchestration ----------------

extern "C" void kernel_launch(void* const* d_in, const int* in_sizes, int n_in,
                              void* d_out, int out_size, void* d_ws, size_t ws_size,
                              hipStream_t stream) {
    (void)in_sizes; (void)n_in; (void)out_size; (void)ws_size;

    const float* x   = (const float*)d_in[0];
    const int* ei    = (const int*)d_in[1];
    const int* srcI  = ei;             // edge_index[0]
    const int* dstI  = ei + N_EDGES;   // edge_index[1]
    const int* batch = (const int*)d_in[2];
    const float* P[32];
    for (int i = 0; i < 32; ++i) P[i] = (const float*)d_in[i + 3];
    // P layout: [layer*6 + {W,b,gamma,beta,mean,var}] for layers init/head/body/tail,
    // then P[24..31] = mlp_W1, mlp_b1, mlp_gamma, mlp_beta, mlp_mean, mlp_var, mlp_W2, mlp_b2

    float* wsf = (float*)d_ws;
    size_t off = 0;
    auto alloc = [&](size_t nf) { float* p = wsf + off; off += (nf + 63) & ~(size_t)63; return p; };
    const size_t NH = (size_t)N_NODES * HID;
    float* deg    = alloc(N_NODES);
    float* nrm    = alloc(N_EDGES);
    float* Bh     = alloc(NH);   // init output; reused for tail output (xt)
    float* Bx1    = alloc(NH);
    float* Bx2    = alloc(NH);
    float* Bx3    = alloc(NH);
    float* T1     = alloc(NH);
    float* T2     = alloc(NH);
    float* T3     = alloc(NH);
    float* pooled = alloc((size_t)NGRAPH * 4 * HID);
    float* z1     = alloc((size_t)NGRAPH * HID);

    const int TB = 256;
    auto nblk = [](long n) { return dim3((unsigned)((n + 255) / 256)); };

    fill_kernel<<<nblk(N_NODES), TB, 0, stream>>>(deg, N_NODES, 0.f);
    deg_kernel<<<nblk(N_EDGES), TB, 0, stream>>>(dstI, deg, N_EDGES);
    norm_kernel<<<nblk(N_EDGES), TB, 0, stream>>>(srcI, dstI, deg, nrm, N_EDGES);

    auto prop = [&](const float* hin, float* tout) {
        fill_kernel<<<nblk((long)NH), TB, 0, stream>>>(tout, (int)NH, 0.f);
        scatter_kernel<<<nblk((long)N_EDGES * 24), TB, 0, stream>>>(srcI, dstI, nrm, hin, tout, N_EDGES);
    };
    auto layer = [&](const float* hin, int pb, float* hout) {
        prop(hin, T1);                                                       // T1 = L h
        prop(T1, T2);
        recur_kernel<<<nblk((long)NH), TB, 0, stream>>>(T2, hin, (int)NH);   // T2 = 2 L T1 - T0
        prop(T2, T3);
        recur_kernel<<<nblk((long)NH), TB, 0, stream>>>(T3, T1, (int)NH);    // T3 = 2 L T2 - T1
        // 50000 rows = 625 blocks * (RB=5 tiles * 16 rows)
        gemm_bn_act_kernel<4, HID, 5><<<dim3(N_NODES / 80), dim3(32, 6), 0, stream>>>(
            hin, T1, T2, T3,
            P[pb + 0], P[pb + 1], P[pb + 4], P[pb + 5], P[pb + 2], P[pb + 3],
            0.01f, hout);
    };

    layer(x,   0,  Bh);    // init
    layer(Bh,  6,  Bx1);   // head
    layer(Bx1, 12, Bx2);   // body
    add_kernel<<<nblk((long)NH), TB, 0, stream>>>(Bx3, Bh, Bx2, (int)NH);  // x3 = h + x2
    layer(Bx3, 18, Bh);    // tail -> xt (reuses Bh)

    fill_kernel<<<nblk(NGRAPH * 4 * HID), TB, 0, stream>>>(pooled, NGRAPH * 4 * HID, -INFINITY);
    pool_kernel<<<nblk((long)N_NODES * 4 * HID), TB, 0, stream>>>(Bh, Bx1, Bx2, Bx3, batch, pooled);

    // z1 = relu(bn(pooled @ W1 + b1)) : 128x384 @ 384x96
    gemm_bn_act_kernel<1, 4 * HID, 1><<<dim3(NGRAPH / 16), dim3(32, 6), 0, stream>>>(
        pooled, nullptr, nullptr, nullptr,
        P[24], P[25], P[28], P[29], P[26], P[27],
        0.0f, z1);

    mlp2_kernel<<<nblk(NGRAPH * NCLS), TB, 0, stream>>>(z1, P[30], P[31], (float*)d_out);
}